// NN_70420283785306
// MI455X (gfx1250) — compile-verified
//
#include <hip/hip_runtime.h>
#include <hip/hip_bf16.h>

typedef __attribute__((ext_vector_type(2))) float v2f;
typedef __attribute__((ext_vector_type(8))) float v8f;

#define WAVES_PER_BLOCK 8
#define STR_Y 10   // y1 staging row stride (floats): even -> 8B-aligned b64 LDS loads
#define STR_H 18   // h  staging row stride (floats): even -> 8B-aligned b64 LDS loads

__device__ __forceinline__ void wave_lds_fence() {
    // Same-wave DS ops are in-order in HW; this only stops the compiler from
    // reordering the cross-lane LDS store->load dependency.
    asm volatile("s_wait_dscnt 0" ::: "memory");
}

__device__ __forceinline__ v8f wmma_f32(v2f a, v2f b, v8f c) {
    return __builtin_amdgcn_wmma_f32_16x16x4_f32(false, a, false, b, (short)0, c,
                                                 false, false);
}

__device__ __forceinline__ float fast_tanh(float x) {
    // 1 - 2/(e^{2x}+1): stable at +/-inf (-> +/-1), v_exp + v_rcp + fma
    float e2 = __expf(2.0f * x);
    return 1.0f - 2.0f * __builtin_amdgcn_rcpf(e2 + 1.0f);
}
__device__ __forceinline__ float fast_sigmoid(float x) {
    return __builtin_amdgcn_rcpf(1.0f + __expf(-x));
}

__global__ __launch_bounds__(256, 2) void moe_wmma_kernel(
    const float* __restrict__ x,  const int* __restrict__ u,
    const float* __restrict__ w1, const float* __restrict__ b1,
    const float* __restrict__ w2, const float* __restrict__ b2,
    const float* __restrict__ w3, const float* __restrict__ b3,
    const float* __restrict__ w4, const float* __restrict__ b4,
    const float* __restrict__ w5, const float* __restrict__ b5,
    const float* __restrict__ w6, const float* __restrict__ b6,
    const float* __restrict__ w7, const float* __restrict__ b7,
    float* __restrict__ out, int nrows)
{
    // -------- per-block weight fragment tables (B-matrix layout, padded) ----
    __shared__ __align__(16) float W1T[16 * 32 * 2];        // [kc][lane][2]
    __shared__ __align__(16) float WAT[3 * 2 * 32 * 2];     // [e][kc][lane][2]
    __shared__ __align__(16) float WBT[3 * 4 * 2 * 32 * 2]; // [e][kc][t][lane][2]
    __shared__ __align__(16) float ybuf_all[WAVES_PER_BLOCK][16 * STR_Y];
    __shared__ __align__(16) float hbuf_all[WAVES_PER_BLOCK][16 * STR_H];

    // Build tables cooperatively (once per block, amortized over 8 tiles).
    for (int i = threadIdx.x; i < 16 * 32; i += 256) {
        int kc = i >> 5, l = i & 31;
        int nn = l & 15, khh = l >> 4;
        int kr = 4 * kc + 2 * khh;
        float m = (nn < 8) ? 1.0f : 0.0f;
        W1T[2 * i + 0] = w1[(kr + 0) * 8 + (nn & 7)] * m;
        W1T[2 * i + 1] = w1[(kr + 1) * 8 + (nn & 7)] * m;
    }
    for (int i = threadIdx.x; i < 3 * 2 * 32; i += 256) {
        int e = i >> 6, kc = (i >> 5) & 1, l = i & 31;
        int nn = l & 15, khh = l >> 4;
        int kr = 4 * kc + 2 * khh;
        const float* wa = (e == 0) ? w2 : ((e == 1) ? w4 : w6);
        WAT[2 * i + 0] = wa[(kr + 0) * 16 + nn];
        WAT[2 * i + 1] = wa[(kr + 1) * 16 + nn];
    }
    for (int i = threadIdx.x; i < 3 * 4 * 2 * 32; i += 256) {
        int e = i >> 8, kc = (i >> 6) & 3, t = (i >> 5) & 1, l = i & 31;
        int nn = (l & 15) + 16 * t, khh = l >> 4;
        int kr = 4 * kc + 2 * khh;
        const float* wb = (e == 0) ? w3 : ((e == 1) ? w5 : w7);
        WBT[2 * i + 0] = wb[(kr + 0) * 32 + nn];
        WBT[2 * i + 1] = wb[(kr + 1) * 32 + nn];
    }
    __syncthreads();

    const int lane = threadIdx.x & 31;
    const int wave = threadIdx.x >> 5;
    const long tile = (long)blockIdx.x * WAVES_PER_BLOCK + wave;
    long base = tile * 16;
    if (base >= nrows) return;          // wave-uniform; EXEC stays all-ones
    // Clamp so every tile is fully in-range (overlap recomputes identical
    // rows -> racing stores write identical bytes; keep 16B alignment for u).
    if (base + 16 > nrows) { base = nrows - 16; if (base < 0) base = 0; base &= ~3L; }

    float* ybuf = ybuf_all[wave];
    float* hbuf = hbuf_all[wave];

    const int n  = lane & 15;           // B/C/D column; also A row (M)
    const int kh = lane >> 4;           // lane-half: K sub-offset selector
    const int k2 = 2 * kh;

    const v8f vzero = {0, 0, 0, 0, 0, 0, 0, 0};

    // ---------------- stage 1: y1 = tanh(x @ w1 - b1)  (N padded 8->16) ----
    const float* xrow = x + (base + n) * 64;
    v2f a1[16];
#pragma unroll
    for (int kc = 0; kc < 16; ++kc)
        a1[kc] = __builtin_nontemporal_load((const v2f*)(xrow + 4 * kc + k2));

    // expert ids for the 8 rows this half-wave owns in D layout (contiguous)
    const int4* up = (const int4*)(u + base + 8 * kh);
    const int4 ua = up[0];
    const int4 ub = up[1];
    int uu[8] = {ua.x, ua.y, ua.z, ua.w, ub.x, ub.y, ub.z, ub.w};

    v8f acc = vzero;                    // C = inline 0; bias folded into tanh
#pragma unroll
    for (int kc = 0; kc < 16; ++kc) {
        v2f bw = *(const v2f*)(W1T + (kc * 32 + lane) * 2);
        acc = wmma_f32(a1[kc], bw, acc);
    }

    // tanh(acc - b1) in D-regs; reg v -> row (v + 8*kh), col n. Stage n<8 only.
    float c1 = b1[n & 7];
#pragma unroll
    for (int v = 0; v < 8; ++v) {
        if (n < 8) ybuf[(v + 8 * kh) * STR_Y + n] = fast_tanh(acc[v] - c1);
    }
    wave_lds_fence();

    // y1 A-fragments (K=8 -> 2 chunks); constant across all experts
    v2f a2[2];
#pragma unroll
    for (int kc = 0; kc < 2; ++kc)
        a2[kc] = *(const v2f*)(ybuf + n * STR_Y + 4 * kc + k2);

    v8f res0 = vzero;
    v8f res1 = vzero;

    auto expert = [&](const float* __restrict__ ba, const float* __restrict__ bb,
                      int eidx) {
        // ---- h = sigmoid(y1 @ wa - ba) : 16x8 @ 8x16, 2 WMMAs ----
        v8f h = vzero;
#pragma unroll
        for (int kc = 0; kc < 2; ++kc) {
            v2f bw = *(const v2f*)(WAT + ((eidx * 2 + kc) * 32 + lane) * 2);
            h = wmma_f32(a2[kc], bw, h);
        }
        float c2 = ba[n];
#pragma unroll
        for (int v = 0; v < 8; ++v) h[v] = fast_sigmoid(h[v] - c2);

        wave_lds_fence();               // WAR vs previous expert's a3 loads
#pragma unroll
        for (int v = 0; v < 8; ++v) hbuf[(v + 8 * kh) * STR_H + n] = h[v];
        wave_lds_fence();

        v2f a3[4];
#pragma unroll
        for (int kc = 0; kc < 4; ++kc)
            a3[kc] = *(const v2f*)(hbuf + n * STR_H + 4 * kc + k2);

        // ---- e = h @ wb - bb : 16x16 @ 16x32 -> two 16x16 N-tiles ----
        v8f e0 = vzero;
        v8f e1 = vzero;
#pragma unroll
        for (int kc = 0; kc < 4; ++kc) {
            v2f bwa = *(const v2f*)(WBT + (((eidx * 4 + kc) * 2 + 0) * 32 + lane) * 2);
            v2f bwb = *(const v2f*)(WBT + (((eidx * 4 + kc) * 2 + 1) * 32 + lane) * 2);
            e0 = wmma_f32(a3[kc], bwa, e0);
            e1 = wmma_f32(a3[kc], bwb, e1);
        }
        // mask-weighted combine (== reference's where-sum), bias folded in
        float c3a = bb[n], c3b = bb[16 + n];
#pragma unroll
        for (int v = 0; v < 8; ++v) {
            float m = (uu[v] == eidx) ? 1.0f : 0.0f;
            res0[v] += m * (e0[v] - c3a);
            res1[v] += m * (e1[v] - c3b);
        }
    };

    expert(b2, b3, 0);
    expert(b4, b5, 1);
    expert(b6, b7, 2);

    // ---- store: D layout -> out[row*32 + col], streamed (NT) ----
#pragma unroll
    for (int v = 0; v < 8; ++v) {
        float* orow = out + (base + v + 8 * kh) * 32;
        __builtin_nontemporal_store(res0[v], orow + n);
        __builtin_nontemporal_store(res1[v], orow + 16 + n);
    }
}

extern "C" void kernel_launch(void* const* d_in, const int* in_sizes, int n_in,
                              void* d_out, int out_size, void* d_ws, size_t ws_size,
                              hipStream_t stream) {
    const float* x  = (const float*)d_in[0];
    const int*   u  = (const int*)d_in[1];
    const float* w1 = (const float*)d_in[2];
    const float* b1 = (const float*)d_in[3];
    const float* w2 = (const float*)d_in[4];
    const float* b2 = (const float*)d_in[5];
    const float* w3 = (const float*)d_in[6];
    const float* b3 = (const float*)d_in[7];
    const float* w4 = (const float*)d_in[8];
    const float* b4 = (const float*)d_in[9];
    const float* w5 = (const float*)d_in[10];
    const float* b5 = (const float*)d_in[11];
    const float* w6 = (const float*)d_in[12];
    const float* b6 = (const float*)d_in[13];
    const float* w7 = (const float*)d_in[14];
    const float* b7 = (const float*)d_in[15];
    float* out = (float*)d_out;

    const int nrows = in_sizes[0] / 64;
    const int tiles = (nrows + 15) / 16;
    const int blocks = (tiles + WAVES_PER_BLOCK - 1) / WAVES_PER_BLOCK;

    hipLaunchKernelGGL(moe_wmma_kernel, dim3(blocks), dim3(256), 0, stream,
                       x, u, w1, b1, w2, b2, w3, b3, w4, b4, w5, b5, w6, b6,
                       w7, b7, out, nrows);
}